// Attention__16381005266984
// MI455X (gfx1250) — compile-verified
//
#include <hip/hip_runtime.h>

// Chunked causal linear attention for MI455X (gfx1250), wave32 + WMMA f32 + TDM.
// B=4 H=8 N=8192 D=64 E=64 C=128 -> nc=64, BH=32, blocks=2048.

typedef float v2f __attribute__((ext_vector_type(2)));
typedef float v8f __attribute__((ext_vector_type(8)));
typedef float f4  __attribute__((ext_vector_type(4)));
typedef unsigned int u32x4 __attribute__((ext_vector_type(4)));
typedef int i32x4 __attribute__((ext_vector_type(4)));
typedef int i32x8 __attribute__((ext_vector_type(8)));

#define WMMA_F32(acc, a, b) \
  (acc) = __builtin_amdgcn_wmma_f32_16x16x4_f32(false, (a), false, (b), (short)0, (acc), false, false)

#if __has_builtin(__builtin_amdgcn_tensor_load_to_lds)
#define USE_TDM 1
#else
#define USE_TDM 0
#endif

static constexpr int kB = 4, kH = 8, kN = 8192, kD = 64, kE = 64, kC = 128;
static constexpr int kNC = kN / kC;          // 64 chunks
static constexpr int kBH = kB * kH;          // 32
static constexpr int kBlocks = kBH * kNC;    // 2048
static constexpr int kDC = 2 * kD;           // 128 (concatenated feature dim)
static constexpr int SQK = kDC + 4;          // 132, padded LDS row stride
static constexpr int SV  = kE + 4;           // 68
static constexpr int SA  = 16 + 4;           // 20
static constexpr float kEps = 1e-6f;

// Workspace layout (floats): wsS[2048][128][64] then wsK[2048][128]
static constexpr size_t kWsS  = (size_t)kBlocks * kDC * kE;   // 16,777,216
// d_out layout (floats): out | Z | S | Z_rot | S_rot
static constexpr size_t kOffZ  = (size_t)kBH * kN * kE;       // 16,777,216
static constexpr size_t kOffS  = kOffZ  + (size_t)kBH * kD;   // +2048
static constexpr size_t kOffZr = kOffS  + (size_t)kBH * kD * kE;
static constexpr size_t kOffSr = kOffZr + (size_t)kBH * kD;

#define LDSOFF(arr) ((uint32_t)(uintptr_t)(void*)(arr))

// Issue one TDM 2-D tile load (global -> LDS), ISA D# per CDNA5 ISA.
// pad_interval_enc: DWORDs-before-pad = 2^(enc+1); pad_dwords: raw DWORD count.
__device__ __forceinline__ void tdm_load_2d(
    uint32_t lds_byte_off, const void* gptr,
    uint32_t tile_w_elems, uint32_t tile_h_rows, uint32_t gstride_elems,
    bool pad_en, uint32_t pad_interval_enc, uint32_t pad_dwords) {
#if USE_TDM
  const uint64_t ga = (uint64_t)(uintptr_t)gptr;
  u32x4 g0;
  g0[0] = 1u;                                    // count=1 (valid), no gather
  g0[1] = lds_byte_off;                          // lds_addr (bytes)
  g0[2] = (uint32_t)ga;                          // global_addr[31:0]
  g0[3] = ((uint32_t)(ga >> 32) & 0x01FFFFFFu)   // global_addr[56:32]
        | (2u << 30);                            // type = 2 ("image")
  i32x8 g1;
  uint32_t dw0 = (2u << 16);                     // data_size = 4 bytes
  if (pad_en)
    dw0 |= (1u << 20) | (pad_interval_enc << 22) | ((pad_dwords - 1u) << 25);
  g1[0] = (int)dw0;                              // wg_mask=0, no atomic barrier
  g1[1] = (int)((tile_w_elems & 0xFFFFu) << 16);                         // tensor_dim0 lo
  g1[2] = (int)((tile_w_elems >> 16) | ((tile_h_rows & 0xFFFFu) << 16)); // td0 hi | tensor_dim1 lo
  g1[3] = (int)((tile_h_rows >> 16) | ((tile_w_elems & 0xFFFFu) << 16)); // td1 hi | tile_dim0
  g1[4] = (int)(tile_h_rows & 0xFFFFu);          // tile_dim1 (tile_dim2 = 0)
  g1[5] = (int)gstride_elems;                    // tensor_dim0_stride[31:0]
  g1[6] = 0;                                     // stride0 hi | stride1 lo
  g1[7] = 0;                                     // stride1 hi
  const i32x4 z4 = {0, 0, 0, 0};
#if __clang_major__ >= 23
  const i32x8 z8 = {0, 0, 0, 0, 0, 0, 0, 0};
  __builtin_amdgcn_tensor_load_to_lds(g0, g1, z4, z4, z8, 0);
#else
  __builtin_amdgcn_tensor_load_to_lds(g0, g1, z4, z4, 0);
#endif
#else
  (void)lds_byte_off; (void)gptr; (void)tile_w_elems; (void)tile_h_rows;
  (void)gstride_elems; (void)pad_en; (void)pad_interval_enc; (void)pad_dwords;
#endif
}

// ---------------- Pass 1: per-chunk local sums  S_loc = Kcat^T V,  ksum = colsum(Kcat)
__global__ __launch_bounds__(256) void lin_attn_pass1(
    const float* __restrict__ k, const float* __restrict__ k_rot,
    const float* __restrict__ v, float* __restrict__ wsS, float* __restrict__ wsK) {
  const int bid  = blockIdx.x;           // bh*NC + c
  const int tid  = threadIdx.x;
  const int lane = tid & 31;
  const int wave = tid >> 5;             // d-tile 0..7
  const int lh   = lane >> 4;            // half
  const int mr   = lane & 15;

  __shared__ float sK[kC * SQK];         // Kcat: 128 x 132
  __shared__ float sVm[kC * SV];         // V:    128 x 68

#if USE_TDM
  if (tid < 32) {
    // row = 64 DWORDs then pad: Kcat halves pad 68 dw (-> stride 132), V pads 4 dw (-> 68)
    tdm_load_2d(LDSOFF(sK),       k     + (size_t)bid * 8192, 64, 128, 64, true, 5, 68);
    tdm_load_2d(LDSOFF(sK) + 256, k_rot + (size_t)bid * 8192, 64, 128, 64, true, 5, 68);
    tdm_load_2d(LDSOFF(sVm),      v     + (size_t)bid * 8192, 64, 128, 64, true, 5, 4);
    __builtin_amdgcn_s_wait_tensorcnt(0);
  }
  __syncthreads();
#else
  const f4* kg  = (const f4*)k     + (size_t)bid * 2048;
  const f4* krg = (const f4*)k_rot + (size_t)bid * 2048;
  const f4* vg  = (const f4*)v     + (size_t)bid * 2048;
  for (int i = tid; i < 2048; i += 256) {
    const int row = i >> 4, c4 = (i & 15) << 2;
    *(f4*)&sK[row * SQK + c4]       = kg[i];
    *(f4*)&sK[row * SQK + 64 + c4]  = krg[i];
    *(f4*)&sVm[row * SV + c4]       = vg[i];
  }
  __syncthreads();
#endif

  const int d0 = 16 * wave;
  v8f acc[4];
#pragma unroll
  for (int te = 0; te < 4; ++te)
#pragma unroll
    for (int r = 0; r < 8; ++r) acc[te][r] = 0.f;

  // S_loc[d][e] = sum_j Kcat[j][d] * V[j][e]; K-dim = 128 -> 32 wmma steps
  for (int ks = 0; ks < 32; ++ks) {
    const int j = 4 * ks + 2 * lh;
    v2f af;
    af.x = sK[j * SQK + d0 + mr];
    af.y = sK[(j + 1) * SQK + d0 + mr];
#pragma unroll
    for (int te = 0; te < 4; ++te) {
      v2f bf;
      bf.x = sVm[j * SV + 16 * te + mr];
      bf.y = sVm[(j + 1) * SV + 16 * te + mr];
      WMMA_F32(acc[te], af, bf);
    }
  }

  float* outS = wsS + (size_t)bid * (kDC * kE);
#pragma unroll
  for (int te = 0; te < 4; ++te)
#pragma unroll
    for (int r = 0; r < 8; ++r)
      outS[(d0 + r + 8 * lh) * kE + 16 * te + mr] = acc[te][r];

  if (tid < kDC) {
    float s = 0.f;
    for (int j = 0; j < kC; ++j) s += sK[j * SQK + tid];
    wsK[(size_t)bid * kDC + tid] = s;
  }
}

// ---------------- Pass 2: exclusive prefix scan across chunks (in-place), emit final states
__global__ __launch_bounds__(256) void lin_attn_pass2(
    float* __restrict__ wsS, float* __restrict__ wsK, float* __restrict__ dout) {
  const int bh  = blockIdx.x;   // 0..31
  const int tid = threadIdx.x;

  float run[32];
#pragma unroll
  for (int i = 0; i < 32; ++i) run[i] = 0.f;

  for (int c = 0; c < kNC; ++c) {
    float* p = wsS + ((size_t)(bh * kNC + c)) * (kDC * kE);
#pragma unroll
    for (int i = 0; i < 32; ++i) {
      const int idx = tid + 256 * i;
      const float val = p[idx];
      p[idx] = run[i];          // exclusive prefix for pass 3
      run[i] += val;
    }
  }
#pragma unroll
  for (int i = 0; i < 32; ++i) {
    const int idx = tid + 256 * i;
    const int d = idx >> 6, e = idx & 63;
    const size_t dst = (d < kD)
        ? (kOffS  + ((size_t)bh * kD + d) * kE + e)
        : (kOffSr + ((size_t)bh * kD + (d - kD)) * kE + e);
    dout[dst] = run[i];
  }

  if (tid < kDC) {
    float runk = 0.f;
    for (int c = 0; c < kNC; ++c) {
      float* p = wsK + ((size_t)(bh * kNC + c)) * kDC + tid;
      const float val = *p;
      *p = runk;
      runk += val;
    }
    const size_t dst = (tid < kD) ? (kOffZ + (size_t)bh * kD + tid)
                                  : (kOffZr + (size_t)bh * kD + (tid - kD));
    dout[dst] = runk;
  }
}

// ---------------- Pass 3: per-chunk output
// A = tril(Qcat Kcat^T); den = rowsum(A) + Qcat.Kprev; out = (A@V + Qcat@Sprev)/(den+eps)
__global__ __launch_bounds__(256) void lin_attn_pass3(
    const float* __restrict__ q, const float* __restrict__ k,
    const float* __restrict__ qr, const float* __restrict__ kr,
    const float* __restrict__ v, const float* __restrict__ wsS,
    const float* __restrict__ wsK, float* __restrict__ out) {
  const int bid  = blockIdx.x;
  const int tid  = threadIdx.x;
  const int lane = tid & 31;
  const int wave = tid >> 5;    // ti: output row-tile 0..7
  const int lh   = lane >> 4;
  const int mr   = lane & 15;

  __shared__ float sQ[kC * SQK];        // Qcat 128x132
  __shared__ float sK[kC * SQK];        // Kcat 128x132
  __shared__ float sVm[kC * SV];        // V    128x68
  __shared__ float sS[kDC * SV];        // Sprev_cat 128x68
  __shared__ float sA[8 * 16 * SA];     // per-wave masked A tile 16x20
  __shared__ float sDen[kC];
  __shared__ float sDinv[kC];
  __shared__ float sKp[kDC];

#if USE_TDM
  if (tid < 32) {
    tdm_load_2d(LDSOFF(sQ),        q   + (size_t)bid * 8192, 64, 128, 64, true, 5, 68);
    tdm_load_2d(LDSOFF(sQ) + 256,  qr  + (size_t)bid * 8192, 64, 128, 64, true, 5, 68);
    tdm_load_2d(LDSOFF(sK),        k   + (size_t)bid * 8192, 64, 128, 64, true, 5, 68);
    tdm_load_2d(LDSOFF(sK) + 256,  kr  + (size_t)bid * 8192, 64, 128, 64, true, 5, 68);
    tdm_load_2d(LDSOFF(sVm),       v   + (size_t)bid * 8192, 64, 128, 64, true, 5, 4);
    tdm_load_2d(LDSOFF(sS),        wsS + (size_t)bid * 8192, 64, 128, 64, true, 5, 4);
    tdm_load_2d(LDSOFF(sKp),       wsK + (size_t)bid * 128, 128, 1, 128, false, 0, 0);
    __builtin_amdgcn_s_wait_tensorcnt(0);
  }
  __syncthreads();
#else
  const f4* qg  = (const f4*)q   + (size_t)bid * 2048;
  const f4* qrg = (const f4*)qr  + (size_t)bid * 2048;
  const f4* kg  = (const f4*)k   + (size_t)bid * 2048;
  const f4* krg = (const f4*)kr  + (size_t)bid * 2048;
  const f4* vg  = (const f4*)v   + (size_t)bid * 2048;
  const f4* sg  = (const f4*)wsS + (size_t)bid * 2048;
  for (int i = tid; i < 2048; i += 256) {
    const int row = i >> 4, c4 = (i & 15) << 2;
    *(f4*)&sQ[row * SQK + c4]      = qg[i];
    *(f4*)&sQ[row * SQK + 64 + c4] = qrg[i];
    *(f4*)&sK[row * SQK + c4]      = kg[i];
    *(f4*)&sK[row * SQK + 64 + c4] = krg[i];
    *(f4*)&sVm[row * SV + c4]      = vg[i];
    *(f4*)&sS[row * SV + c4]       = sg[i];
  }
  if (tid < kDC) sKp[tid] = wsK[(size_t)bid * kDC + tid];
  __syncthreads();
#endif

  const int ti = wave;
  float* sAw = &sA[wave * 16 * SA];

  v8f acc2[4];
  float den[8];
#pragma unroll
  for (int te = 0; te < 4; ++te)
#pragma unroll
    for (int r = 0; r < 8; ++r) acc2[te][r] = 0.f;
#pragma unroll
  for (int r = 0; r < 8; ++r) den[r] = 0.f;

  for (int tj = 0; tj <= ti; ++tj) {
    // --- Atile = Qcat[ti-rows] . Kcat[tj-rows]^T  (K-dim 128, 32 steps)
    v8f a;
#pragma unroll
    for (int r = 0; r < 8; ++r) a[r] = 0.f;
    for (int ks = 0; ks < 32; ++ks) {
      const int kk = 4 * ks + 2 * lh;
      v2f af, bf;
      af.x = sQ[(16 * ti + mr) * SQK + kk];
      af.y = sQ[(16 * ti + mr) * SQK + kk + 1];
      bf.x = sK[(16 * tj + mr) * SQK + kk];
      bf.y = sK[(16 * tj + mr) * SQK + kk + 1];
      WMMA_F32(a, af, bf);
    }
    if (tj == ti) {            // causal mask (keep n <= m), diagonal inclusive
#pragma unroll
      for (int r = 0; r < 8; ++r)
        if (mr > r + 8 * lh) a[r] = 0.f;
    }
    // --- rowsum into den (reduce across 16-lane half), stage Atile to LDS
#pragma unroll
    for (int r = 0; r < 8; ++r) {
      float vv = a[r];
      vv += __shfl_xor(vv, 1);
      vv += __shfl_xor(vv, 2);
      vv += __shfl_xor(vv, 4);
      vv += __shfl_xor(vv, 8);
      den[r] += vv;
      sAw[(r + 8 * lh) * SA + mr] = a[r];
    }
    __builtin_amdgcn_wave_barrier();   // order LDS store -> load within wave
    // --- acc2 += Atile @ V[tj-rows]  (K-dim 16, 4 steps per output tile)
#pragma unroll
    for (int te = 0; te < 4; ++te) {
#pragma unroll
      for (int js = 0; js < 4; ++js) {
        const int jj = 4 * js + 2 * lh;
        v2f af, bf;
        af.x = sAw[mr * SA + jj];
        af.y = sAw[mr * SA + jj + 1];
        bf.x = sVm[(16 * tj + jj) * SV + 16 * te + mr];
        bf.y = sVm[(16 * tj + jj + 1) * SV + 16 * te + mr];
        WMMA_F32(acc2[te], af, bf);
      }
    }
    __builtin_amdgcn_wave_barrier();
  }

  // --- acc2 += Qcat[ti-rows] @ Sprev_cat (K-dim 128, 32 steps)
  for (int ks = 0; ks < 32; ++ks) {
    const int kk = 4 * ks + 2 * lh;
    v2f af;
    af.x = sQ[(16 * ti + mr) * SQK + kk];
    af.y = sQ[(16 * ti + mr) * SQK + kk + 1];
#pragma unroll
    for (int te = 0; te < 4; ++te) {
      v2f bf;
      bf.x = sS[kk * SV + 16 * te + mr];
      bf.y = sS[(kk + 1) * SV + 16 * te + mr];
      WMMA_F32(acc2[te], af, bf);
    }
  }

  if (mr == 0) {
#pragma unroll
    for (int r = 0; r < 8; ++r) sDen[16 * ti + r + 8 * lh] = den[r];
  }
  __syncthreads();
  if (tid < kC) {
    float dot = 0.f;
    for (int dd = 0; dd < kDC; ++dd) dot += sQ[tid * SQK + dd] * sKp[dd];
    sDinv[tid] = 1.0f / (sDen[tid] + dot + kEps);
  }
  __syncthreads();

  float dinv[8];
#pragma unroll
  for (int r = 0; r < 8; ++r) dinv[r] = sDinv[16 * ti + r + 8 * lh];

  float* og = out + (size_t)bid * (kC * kE);
#pragma unroll
  for (int te = 0; te < 4; ++te)
#pragma unroll
    for (int r = 0; r < 8; ++r)
      og[(16 * ti + r + 8 * lh) * kE + 16 * te + mr] = acc2[te][r] * dinv[r];
}

extern "C" void kernel_launch(void* const* d_in, const int* in_sizes, int n_in,
                              void* d_out, int out_size, void* d_ws, size_t ws_size,
                              hipStream_t stream) {
  const float* q     = (const float*)d_in[0];
  const float* k     = (const float*)d_in[1];
  const float* q_rot = (const float*)d_in[2];
  const float* k_rot = (const float*)d_in[3];
  const float* v     = (const float*)d_in[4];
  float* out = (float*)d_out;

  float* wsS = (float*)d_ws;
  float* wsK = wsS + kWsS;
  (void)in_sizes; (void)n_in; (void)out_size; (void)ws_size;

  lin_attn_pass1<<<kBlocks, 256, 0, stream>>>(k, k_rot, v, wsS, wsK);
  lin_attn_pass2<<<kBH, 256, 0, stream>>>(wsS, wsK, out);
  lin_attn_pass3<<<kBlocks, 256, 0, stream>>>(q, k, q_rot, k_rot, v, wsS, wsK, out);
}